// voting_56478819942640
// MI455X (gfx1250) — compile-verified
//
#include <hip/hip_runtime.h>

#define ALPHA 1.0f

typedef __attribute__((ext_vector_type(2))) float v2f;
typedef __attribute__((ext_vector_type(8))) float v8f;

// ---------------------------------------------------------------------------
// Stage A: s[b][n] = sum_t spikes[b][t][n]   (pure streaming, 655 MB read)
// float4 per thread, fully coalesced; grid sized to saturate 23.3 TB/s HBM.
// ---------------------------------------------------------------------------
__global__ __launch_bounds__(256) void spike_sum_kernel(
    const float* __restrict__ spikes, float* __restrict__ s,
    int B, int T, int N)
{
    const int n4 = N >> 2;                       // float4 columns per row
    const int total = B * n4;
    int i = blockIdx.x * blockDim.x + threadIdx.x;
    if (i >= total) return;
    int b = i / n4;
    int j = i - b * n4;

    const float4* __restrict__ src =
        (const float4*)spikes + (size_t)b * T * n4 + j;
    float4 acc = {0.f, 0.f, 0.f, 0.f};
    for (int t = 0; t < T; ++t) {
        if (t + 8 < T)
            __builtin_prefetch(src + (size_t)(t + 8) * n4, 0, 1); // global_prefetch_b8
        float4 v = src[(size_t)t * n4];
        acc.x += v.x; acc.y += v.y; acc.z += v.z; acc.w += v.w;
    }
    ((float4*)s)[(size_t)b * n4 + j] = acc;
}

// ---------------------------------------------------------------------------
// Stage B: seg_sum = onehot(labels)^T [16 x B] @ s [B x N] via
// V_WMMA_F32_16X16X4_F32, then fused finalize (mean, rates_new, argmax).
// One wave (32 threads) per 16-wide neuron tile.
//
// A layout (16x4 f32): lanes 0-15: M=lane, {VGPR0:K=0, VGPR1:K=1}
//                      lanes16-31: M=lane-16, {VGPR0:K=2, VGPR1:K=3}
// B layout (4x16 f32): lanes 0-15: N=lane, {VGPR0:K=0, VGPR1:K=1}
//                      lanes16-31: N=lane-16,{VGPR0:K=2, VGPR1:K=3}
// C layout (16x16 f32): VGPR v: row = v + 8*(lane>=16), col = lane&15
// ---------------------------------------------------------------------------
__global__ __launch_bounds__(32) void seg_wmma_kernel(
    const float* __restrict__ s, const int* __restrict__ labels,
    const float* __restrict__ rates, float* __restrict__ out,
    int B, int N, int L)
{
    __shared__ int lbl[1024];
    const int lane  = threadIdx.x;          // 0..31 (wave32)
    const int m     = lane & 15;            // row (A) / col (B,C) within tile
    const int khalf = lane >> 4;            // 0 or 1
    const int k0    = khalf << 1;           // 0 or 2

    for (int i = lane; i < B; i += 32) lbl[i] = labels[i];
    __syncthreads();

    const int n0   = blockIdx.x * 16;
    const int ncol = n0 + m;

    v8f c = {};
    const float* __restrict__ sp = s + (size_t)k0 * N + n0 + m;
    for (int k = 0; k < B; k += 4) {
        v2f a, bm;
        a.x  = (lbl[k + k0]     == m) ? 1.0f : 0.0f;
        a.y  = (lbl[k + k0 + 1] == m) ? 1.0f : 0.0f;
        bm.x = sp[0];
        bm.y = sp[N];
        // D = A x B + C ; fp32 in / fp32 accumulate -> exact segment sums
        c = __builtin_amdgcn_wmma_f32_16x16x4_f32(
                /*neg_a=*/false, a, /*neg_b=*/false, bm,
                /*c_mod=*/(short)0, c, /*reuse_a=*/false, /*reuse_b=*/false);
        sp += (size_t)4 * N;
    }

    // counts: lane l (0..15) counts occurrences of label l
    int cnt_m = 0;
    for (int i = 0; i < B; ++i) cnt_m += (lbl[i] == m) ? 1 : 0;

    // finalize: rates_new[n,l] = cnt[l]>0 ? ALPHA*rates[n,l] + seg/cnt : rates[n,l]
    float bestv = -3.402823466e38f;
    int   bestl = 0x7fffffff;
    float* __restrict__ rates_out = out + N;   // out[0..N) = assignments
    for (int v = 0; v < 8; ++v) {
        int l = v + 8 * khalf;
        int cl = __shfl(cnt_m, l, 32);         // counts[l] lives in lane l
        if (l < L) {
            float r    = rates[(size_t)ncol * L + l];
            float mean = c[v] / fmaxf((float)cl, 1.0f);
            float rn   = (cl > 0) ? (ALPHA * r + mean) : r;
            rates_out[(size_t)ncol * L + l] = rn;
            if (rn > bestv) { bestv = rn; bestl = l; }   // first-max tie-break
        }
    }

    // combine the two lane halves holding column ncol (lanes n and n+16)
    float ov = __shfl_xor(bestv, 16, 32);
    int   ol = __shfl_xor(bestl, 16, 32);
    if (ov > bestv || (ov == bestv && ol < bestl)) { bestv = ov; bestl = ol; }
    if (khalf == 0) out[ncol] = (float)bestl;  // assignments as float dtype
}

// ---------------------------------------------------------------------------
extern "C" void kernel_launch(void* const* d_in, const int* in_sizes, int n_in,
                              void* d_out, int out_size, void* d_ws, size_t ws_size,
                              hipStream_t stream)
{
    const float* spikes = (const float*)d_in[0];   // [B,T,N] f32
    const int*   labels = (const int*)  d_in[1];   // [B]     i32
    const float* rates  = (const float*)d_in[2];   // [N,L]   f32
    float*       out    = (float*)d_out;           // [N] + [N,L]

    const int B = in_sizes[1];                     // 512
    const int N = out_size - in_sizes[2];          // 70400 - 64000 = 6400
    const int L = in_sizes[2] / N;                 // 10
    const int T = in_sizes[0] / (B * N);           // 50

    float* s = (float*)d_ws;                       // [B,N] scratch (13.1 MB)

    // Stage A: bandwidth-bound T-reduction
    {
        int total  = B * (N >> 2);
        int blocks = (total + 255) / 256;
        spike_sum_kernel<<<blocks, 256, 0, stream>>>(spikes, s, B, T, N);
    }
    // Stage B: WMMA segment-sum + fused finalize (one wave per 16 neurons)
    {
        int tiles = (N + 15) / 16;                 // 400
        seg_wmma_kernel<<<tiles, 32, 0, stream>>>(s, labels, rates, out, B, N, L);
    }
}